// LengthRegulator_6356551598478
// MI455X (gfx1250) — compile-verified
//
#include <hip/hip_runtime.h>
#include <stdint.h>

#define B_      32
#define T_      1024
#define D_      512
#define MAXLEN  8192
#define PB      32      // output positions handled per gather block

typedef float v4f __attribute__((ext_vector_type(4)));

// ---------------------------------------------------------------------------
// Kernel 1: per-batch inclusive scan of duration (int64, values 0..7 -> int32)
// One 1024-thread block per batch. Hillis-Steele in LDS.
// ---------------------------------------------------------------------------
__global__ __launch_bounds__(1024)
void lr_scan_kernel(const long long* __restrict__ dur,
                    int* __restrict__ cum,
                    long long* __restrict__ mel_len)
{
    __shared__ int s[T_];
    const int b = blockIdx.x;
    const int t = threadIdx.x;

    s[t] = (int)dur[(size_t)b * T_ + t];
    __syncthreads();

    #pragma unroll
    for (int off = 1; off < T_; off <<= 1) {
        int add = (t >= off) ? s[t - off] : 0;
        __syncthreads();
        s[t] += add;
        __syncthreads();
    }

    cum[b * T_ + t] = s[t];
    if (t == T_ - 1) mel_len[b] = (long long)s[t];
}

// ---------------------------------------------------------------------------
// Kernel 2: searchsorted + masked row gather.
// grid = (MAXLEN/PB, B), block = 256 threads (8 waves).
//  - stage cum[b][0..1023] (4 KB) into LDS via async-to-LDS (16B per lane)
//  - 32 threads binary-search their position's index
//  - 2 x 128-thread groups stream one 2KB row per iteration (b128 ld/st)
//  - output stores are NON-TEMPORAL: 512 MB write-once stream must not
//    evict the reusable 64 MB x working set from the 192 MB L2
// ---------------------------------------------------------------------------
__global__ __launch_bounds__(256)
void lr_gather_kernel(const float* __restrict__ x,
                      const int* __restrict__ cum,
                      float* __restrict__ out)
{
    __shared__ int cs[T_];
    __shared__ int idxs[PB];

    const int b   = blockIdx.y;
    const int tid = threadIdx.x;

    // --- async stage of cum row into LDS: 256 lanes x 16B = 4096B exactly ---
    {
        unsigned lds_off = (unsigned)(uintptr_t)(&cs[tid * 4]);
        uint64_t gaddr   = (uint64_t)(uintptr_t)(cum + b * T_ + tid * 4);
        asm volatile("global_load_async_to_lds_b128 %0, %1, off"
                     :: "v"(lds_off), "v"(gaddr) : "memory");
    }
    asm volatile("s_wait_asynccnt 0" ::: "memory");
    __syncthreads();

    // --- per-position index: count of cum[j] <= p (searchsorted 'right') ---
    if (tid < PB) {
        int p  = blockIdx.x * PB + tid;
        int lo = 0, hi = T_;
        #pragma unroll
        for (int step = 0; step < 10; ++step) {   // log2(1024) iterations
            int mid = (lo + hi) >> 1;
            if (cs[mid] <= p) lo = mid + 1; else hi = mid;
        }
        int idx = (lo < T_) ? lo : (T_ - 1);
        idxs[tid] = (p < cs[T_ - 1]) ? idx : -1;  // -1 => masked (zero row)
    }
    __syncthreads();

    // --- row copy: 128 threads x float4 = 2048B = one full D-row ---
    const int half = tid >> 7;     // which of 2 rows this pass
    const int lane = tid & 127;    // float4 slot within the row

    #pragma unroll 4
    for (int it = 0; it < PB / 2; ++it) {
        const int pl  = it * 2 + half;
        const int p   = blockIdx.x * PB + pl;
        const int idx = idxs[pl];

        v4f* dst = (v4f*)(out + ((size_t)b * MAXLEN + p) * D_) + lane;
        if (idx >= 0) {
            const v4f* src = (const v4f*)(x + ((size_t)b * T_ + idx) * D_) + lane;
            v4f v = *src;                       // temporal: keep x hot in L2
            __builtin_nontemporal_store(v, dst); // NT: stream out, bypass reuse
        } else {
            v4f z = (v4f)0.0f;
            __builtin_nontemporal_store(z, dst);
        }
    }
}

// ---------------------------------------------------------------------------
// Launch
// ---------------------------------------------------------------------------
extern "C" void kernel_launch(void* const* d_in, const int* in_sizes, int n_in,
                              void* d_out, int out_size, void* d_ws, size_t ws_size,
                              hipStream_t stream)
{
    const float*     x   = (const float*)d_in[0];
    const long long* dur = (const long long*)d_in[1];
    (void)in_sizes; (void)n_in; (void)out_size; (void)ws_size;

    float* out = (float*)d_out;
    // mel_len (int64) lives after the float payload, flat in return order
    long long* mel = (long long*)(out + (size_t)B_ * MAXLEN * D_);
    int* cum = (int*)d_ws;  // 32*1024*4 = 128 KB scratch

    lr_scan_kernel<<<B_, T_, 0, stream>>>(dur, cum, mel);
    lr_gather_kernel<<<dim3(MAXLEN / PB, B_), 256, 0, stream>>>(x, cum, out);
}